// TTCLoss_62105227100263
// MI455X (gfx1250) — compile-verified
//
#include <hip/hip_runtime.h>
#include <float.h>
#include <math.h>

#define TIME_INTERVAL 0.5f

// Native clang vector type: __builtin_nontemporal_load requires scalar or
// native vector pointers (HIP's float4 class is rejected).
typedef float nfloat4 __attribute__((ext_vector_type(4)));

// Per-element loss term, matching the JAX reference:
//  preds:   clamp [0, 0.5] -> 0.6 (NaN preds propagate: compares are false)
//  targets: clamp [0, 0.5] or NaN -> 0.6
//  term = | log(1 - ti/p) - log(1 - ti/t) |
// Direct AMDGCN builtins (HIP's __fdividef/__logf only go native under
// -ffast-math, which we don't control): v_rcp_f32 + fma + v_log_f32 per side,
// ~1 ulp each, log args are normal-range (>= ~1e-7). ~10 VALU ops/element vs
// the ~25-op budget at 23.3 TB/s keeps this kernel HBM-bound.
__device__ __forceinline__ float ttc_term(float p, float t) {
    p = ((p >= 0.0f) && (p <= 0.5f)) ? 0.6f : p;
    t = (((t >= 0.0f) && (t <= 0.5f)) || __builtin_isnan(t)) ? 0.6f : t;
    float lp = __builtin_amdgcn_logf(1.0f - TIME_INTERVAL * __builtin_amdgcn_rcpf(p));
    float lt = __builtin_amdgcn_logf(1.0f - TIME_INTERVAL * __builtin_amdgcn_rcpf(t));
    return fabsf(lp - lt);
}

// wave32 + LDS block reduction (gfx1250 is wave32-only: 5 xor steps per wave).
// Result valid for threadIdx.x == 0.
__device__ __forceinline__ float block_reduce_sum(float v) {
    #pragma unroll
    for (int off = 16; off > 0; off >>= 1)
        v += __shfl_xor(v, off, 32);

    __shared__ float wave_sums[32];            // supports up to 1024 threads
    const int lane = threadIdx.x & 31;
    const int wave = threadIdx.x >> 5;
    if (lane == 0) wave_sums[wave] = v;
    __syncthreads();

    if (wave == 0) {
        const int nwaves = (blockDim.x + 31) >> 5;
        v = (lane < nwaves) ? wave_sums[lane] : 0.0f;
        #pragma unroll
        for (int off = 16; off > 0; off >>= 1)
            v += __shfl_xor(v, off, 32);
    }
    return v;
}

// Pass 1: grid-stride streaming over float4 (global_load_b128 th:TH_LOAD_NT),
// non-temporal since each byte is touched once (268 MB total vs 192 MB L2).
__global__ void ttc_partial_kernel(const float* __restrict__ preds,
                                   const float* __restrict__ tgts,
                                   float* __restrict__ partial,
                                   int nvec, int n) {
    const nfloat4* __restrict__ p4 = reinterpret_cast<const nfloat4*>(preds);
    const nfloat4* __restrict__ t4 = reinterpret_cast<const nfloat4*>(tgts);

    float acc = 0.0f;
    const int stride = gridDim.x * blockDim.x;
    const int tid    = blockIdx.x * blockDim.x + threadIdx.x;

    for (int i = tid; i < nvec; i += stride) {
        nfloat4 p = __builtin_nontemporal_load(&p4[i]);
        nfloat4 t = __builtin_nontemporal_load(&t4[i]);
        acc += ttc_term(p.x, t.x);
        acc += ttc_term(p.y, t.y);
        acc += ttc_term(p.z, t.z);
        acc += ttc_term(p.w, t.w);
    }
    // Scalar tail for n not divisible by 4 (n = 2^25 has no tail, but be safe).
    for (int i = 4 * nvec + tid; i < n; i += stride) {
        float p = __builtin_nontemporal_load(&preds[i]);
        float t = __builtin_nontemporal_load(&tgts[i]);
        acc += ttc_term(p, t);
    }

    float s = block_reduce_sum(acc);
    if (threadIdx.x == 0) partial[blockIdx.x] = s;
}

// Pass 2: single block reduces the per-block partials (deterministic tree,
// no float atomics) and applies the avg_factor normalization.
__global__ void ttc_final_kernel(const float* __restrict__ partial, int nblocks,
                                 const int* __restrict__ avg_factor,
                                 float* __restrict__ out) {
    float acc = 0.0f;
    for (int i = threadIdx.x; i < nblocks; i += blockDim.x)
        acc += partial[i];
    float s = block_reduce_sum(acc);
    if (threadIdx.x == 0) {
        float denom = (float)(*avg_factor) + FLT_EPSILON;  // matches np.float32 eps
        out[0] = s / denom;
    }
}

extern "C" void kernel_launch(void* const* d_in, const int* in_sizes, int n_in,
                              void* d_out, int out_size, void* d_ws, size_t ws_size,
                              hipStream_t stream) {
    const float* preds = (const float*)d_in[0];
    const float* tgts  = (const float*)d_in[1];
    const int*   avg   = (const int*)d_in[2];   // python int scalar -> 1-elem int array
    float*       out   = (float*)d_out;
    float*       part  = (float*)d_ws;

    const int n       = in_sizes[0];
    const int nvec    = n / 4;
    const int threads = 256;                     // 8 wave32 per block

    int blocks = (nvec + threads - 1) / threads;
    if (blocks > 2048) blocks = 2048;            // 2048 partials = 8 KB of d_ws
    int ws_cap = (int)(ws_size / sizeof(float));
    if (blocks > ws_cap) blocks = ws_cap;
    if (blocks < 1) blocks = 1;

    ttc_partial_kernel<<<blocks, threads, 0, stream>>>(preds, tgts, part, nvec, n);
    ttc_final_kernel<<<1, 256, 0, stream>>>(part, blocks, avg, out);
}